// WaveletDecomposer1D_4681514352822
// MI455X (gfx1250) — compile-verified
//
#include <hip/hip_runtime.h>

typedef float v2f __attribute__((ext_vector_type(2)));
typedef float v4f __attribute__((ext_vector_type(4)));
typedef float v8f __attribute__((ext_vector_type(8)));

// Haar 3-level block coefficients: w(t, s) = weight of sample s (0..7) for
// coefficient type t within an 8-sample block.
//   t=0: cA3, t=1: cD3, t=2: cD2[0], t=3: cD2[1], t=4..7: cD1[0..3]
__device__ __forceinline__ float wcoef(int t, int s) {
  const float S1 = 0.70710678118654752440f;   // 2^-1/2
  const float S2 = 0.5f;                      // 2^-1
  const float S3 = 0.35355339059327376220f;   // 2^-3/2
  switch (t) {
    case 0: return S3;
    case 1: return (s < 4) ? S3 : -S3;
    case 2: return (s < 2) ? S2 : ((s < 4) ? -S2 : 0.0f);
    case 3: return (s < 4) ? 0.0f : ((s < 6) ? S2 : -S2);
    default: {
      int j = t - 4;
      return (s == 2 * j) ? S1 : ((s == 2 * j + 1) ? -S1 : 0.0f);
    }
  }
}

// One wave processes one 256-sample segment per loop iteration:
// 16 columns (groups) of 16 samples = 32 Haar blocks of 8 samples.
// D(16x16) = W16(16x16) x X(16x16) via 4 chained V_WMMA_F32_16X16X4_F32.
__global__ __launch_bounds__(256) void haar3_wmma_kernel(
    const float* __restrict__ x, float* __restrict__ out, int nrows) {
  const int L = 16384;                 // row length (fixed by reference)
  const int lane = threadIdx.x & 31;
  const int wave = threadIdx.x >> 5;
  const int wavesPerBlock = blockDim.x >> 5;
  const int waveGid = blockIdx.x * wavesPerBlock + wave;
  const int nWaves  = gridDim.x * wavesPerBlock;
  const int segPerRow = L / 256;       // 64
  const int nseg = nrows * segPerRow;

  const int hi = lane >> 4;            // 0: K-half one (samples 0..7), 1: samples 8..15
  const int g  = lane & 15;            // column N (= group), and row index M for A
  const int t  = lane & 7;             // coefficient type of this lane's A rows
  // A-lane is "active" when its K-half (sample sub-block) matches its row's
  // sub-block in blockdiag(W8, W8): bit4(lane) == bit3(lane).
  const bool active = (((lane >> 4) & 1) == ((lane >> 3) & 1));

  // Constant A operand (W16 with K permuted so lanes 0-15 <-> samples 0..7,
  // lanes 16-31 <-> samples 8..15). Chunk c covers samples (2c, 2c+1) of the
  // lane's sub-block.
  v2f a0, a1, a2, a3;
  a0.x = active ? wcoef(t, 0) : 0.0f;  a0.y = active ? wcoef(t, 1) : 0.0f;
  a1.x = active ? wcoef(t, 2) : 0.0f;  a1.y = active ? wcoef(t, 3) : 0.0f;
  a2.x = active ? wcoef(t, 4) : 0.0f;  a2.y = active ? wcoef(t, 5) : 0.0f;
  a3.x = active ? wcoef(t, 6) : 0.0f;  a3.y = active ? wcoef(t, 7) : 0.0f;

  float* outA  = out;                               // cA3: nrows*2048
  float* outD3 = out + (size_t)nrows * 2048;        // cD3: nrows*2048
  float* outD2 = out + (size_t)nrows * 4096;        // cD2: nrows*4096
  float* outD1 = out + (size_t)nrows * 8192;        // cD1: nrows*8192

  for (int s = waveGid; s < nseg; s += nWaves) {
    const int b   = s >> 6;            // row (batch) index
    const int seg = s & 63;            // segment within row

    // Lane loads its 8 samples: sub-block `hi` of group `g` (32 contiguous B).
    const float* p = x + (size_t)s * 256 + g * 16 + hi * 8;
    v4f lo4 = *(const v4f*)(p);        // samples +0..3  -> chunks 0,1
    v4f hi4 = *(const v4f*)(p + 4);    // samples +4..7  -> chunks 2,3
    __builtin_prefetch(p + (size_t)nWaves * 256, 0, 1);

    v2f b0 = {lo4.x, lo4.y};
    v2f b1 = {lo4.z, lo4.w};
    v2f b2 = {hi4.x, hi4.y};
    v2f b3 = {hi4.z, hi4.w};

    v8f acc = {};
    acc = __builtin_amdgcn_wmma_f32_16x16x4_f32(false, a0, false, b0,
                                                (short)0, acc, false, false);
    acc = __builtin_amdgcn_wmma_f32_16x16x4_f32(false, a1, false, b1,
                                                (short)0, acc, false, false);
    acc = __builtin_amdgcn_wmma_f32_16x16x4_f32(false, a2, false, b2,
                                                (short)0, acc, false, false);
    acc = __builtin_amdgcn_wmma_f32_16x16x4_f32(false, a3, false, b3,
                                                (short)0, acc, false, false);

    // acc[j] = coefficient type j of this lane's block.
    // Block within row: lanes 0-15 -> even blocks 2g, lanes 16-31 -> odd 2g+1.
    const size_t blk = (size_t)seg * 32 + 2 * g + hi;   // 0..2047 within row
    outA [(size_t)b * 2048 + blk] = acc[0];
    outD3[(size_t)b * 2048 + blk] = acc[1];
    v2f d2v = {acc[2], acc[3]};
    *(v2f*)(outD2 + (size_t)b * 4096 + 2 * blk) = d2v;
    v4f d1v = {acc[4], acc[5], acc[6], acc[7]};
    *(v4f*)(outD1 + (size_t)b * 8192 + 4 * blk) = d1v;
  }
}

extern "C" void kernel_launch(void* const* d_in, const int* in_sizes, int n_in,
                              void* d_out, int out_size, void* d_ws, size_t ws_size,
                              hipStream_t stream) {
  (void)n_in; (void)out_size; (void)d_ws; (void)ws_size;
  const float* x = (const float*)d_in[0];
  float* out = (float*)d_out;
  const int L = 16384;
  const int nrows = in_sizes[0] / L;           // 2048
  const int nseg = nrows * (L / 256);          // 131072 wave-segments

  const int threads = 256;                     // 8 waves per block
  int blocks = (nseg + 8 - 1) / 8 / 4;         // ~4 segments per wave
  if (blocks < 1) blocks = 1;
  if (blocks > 16384) blocks = 16384;
  haar3_wmma_kernel<<<blocks, threads, 0, stream>>>(x, out, nrows);
}